// OptimizedPositionalGAT_51196010169101
// MI455X (gfx1250) — compile-verified
//
#include <hip/hip_runtime.h>
#include <hip/hip_bf16.h>
#include <math.h>

// ---------------- problem constants (match reference) ----------------
#define NN      50000
#define EE      800000
#define ETOT    (EE + NN)      // edges + self loops
#define BB      64
#define F_IND   128
#define POSD    16
#define ENHD    144            // F_IN + POS
#define HEADS   4
#define CHD     64
#define HCD     256            // HEADS*CHD
#define OUTD    128
#define EPSBN   1e-5f
#define NSLOPE  0.2f

typedef float v2f __attribute__((ext_vector_type(2)));
typedef float v8f __attribute__((ext_vector_type(8)));

// ---------------- small helpers ----------------
__device__ __forceinline__ void atomicMaxF(float* addr, float val) {
    unsigned* ua = (unsigned*)addr;
    unsigned old = *ua;
    while (__uint_as_float(old) < val) {
        unsigned assumed = old;
        old = atomicCAS(ua, assumed, __float_as_uint(val));
        if (old == assumed) break;
    }
}

__device__ __forceinline__ void edge_sd(const int* __restrict__ ei, int e, int& s, int& d) {
    if (e < EE) { s = ei[e]; d = ei[EE + e]; }
    else        { s = e - EE; d = s; }       // self loop
}

// ---------------- utility kernels ----------------
__global__ void fill_f32(float* __restrict__ p, float v, int n) {
    int i = blockIdx.x * blockDim.x + threadIdx.x;
    if (i < n) p[i] = v;
}

__global__ void zero_i32(int* __restrict__ p, int n) {
    int i = blockIdx.x * blockDim.x + threadIdx.x;
    if (i < n) p[i] = 0;
}

// ---------------- graph meta ----------------
__global__ void count_nodes(const int* __restrict__ batch, int* __restrict__ counts) {
    int i = blockIdx.x * blockDim.x + threadIdx.x;
    if (i < NN) atomicAdd(&counts[batch[i]], 1);
}

__global__ void graph_meta(const int* __restrict__ counts, int* __restrict__ starts,
                           int* __restrict__ gsz) {
    if (threadIdx.x == 0 && blockIdx.x == 0) {
        int acc = 0;
        for (int g = 0; g < BB; ++g) {
            starts[g] = acc;
            int c = counts[g];
            acc += c;
            int gs = (int)ceilf(sqrtf((float)c));
            gsz[g] = gs > 0 ? gs : 1;
        }
    }
}

// h0 = [x | pos @ W_pos + b_pos],  row stride ENHD
__global__ void build_h0(const float* __restrict__ x, const int* __restrict__ batch,
                         const int* __restrict__ starts, const int* __restrict__ gsz,
                         const float* __restrict__ W_pos, const float* __restrict__ b_pos,
                         float* __restrict__ h0) {
    int idx = blockIdx.x * blockDim.x + threadIdx.x;
    if (idx >= NN * ENHD) return;
    int n = idx / ENHD, j = idx - n * ENHD;
    if (j < F_IND) {
        h0[idx] = x[(size_t)n * F_IND + j];
    } else {
        int jj = j - F_IND;
        int b  = batch[n];
        int i  = n - starts[b];
        int g  = gsz[b];
        int row = i / g, col = i - row * g;
        float dn = fmaxf((float)(g - 1), 1.0f);
        float p0 = (float)row / dn, p1 = (float)col / dn;
        h0[idx] = p0 * W_pos[jj] + p1 * W_pos[POSD + jj] + b_pos[jj];
    }
}

// ---------------- WMMA f32 GEMM (K compile-time): C = A[M,K] @ Bw[K,ncols] (+bias) ----
// block = 128 threads (4 waves). Block computes 64 rows x 64 cols.
// B panel (K x 64) staged in LDS as interleaved k-row PAIRS:
//   element (k,c) at ldsB[(k>>1)*LPAIR + 2*c + (k&1)]
// so each B fragment {B[kb][c], B[kb+1][c]} (kb always even) is ONE aligned
// ds_load_b64 straight into an even VGPR pair -> no register shuffles.
// LPAIR = 160 dwords: lanes 0-15 (pair j) vs 16-31 (pair j+1) differ by
// 160 mod 64 = 32 banks -> the two wave halves hit disjoint bank sets.
// K-loop fully unrolled; A-fragment prefetched one step ahead.
template <int K>
__global__ void gemm_wmma_f32_t(const float* __restrict__ A, const float* __restrict__ Bw,
                                const float* __restrict__ bias, float* __restrict__ C,
                                int M, int ncols) {
    constexpr int LPAIR = 160;               // dword stride per k-row pair (128 data + 32 pad)
    __shared__ float ldsB[(K / 2) * LPAIR];

    const int tid = threadIdx.x;
    const int colBase = blockIdx.y * 64;

    // ---- cooperative stage of B panel [K x 64] into pair-interleaved LDS ----
    {
        const int jrp = tid >> 4;            // row-pair 0..7 within a pass
        const int cc  = (tid & 15) * 4;      // 0,4,...,60
#pragma unroll
        for (int r0 = 0; r0 < K; r0 += 16) { // 8 row-pairs per pass
            const int j = (r0 >> 1) + jrp;   // global row-pair index
            const int r = 2 * j;
            const float4 v0 = *(const float4*)(Bw + (size_t)r * ncols + colBase + cc);
            const float4 v1 = *(const float4*)(Bw + (size_t)(r + 1) * ncols + colBase + cc);
            float4 w0, w1;                   // interleave rows r, r+1
            w0.x = v0.x; w0.y = v1.x; w0.z = v0.y; w0.w = v1.y;
            w1.x = v0.z; w1.y = v1.z; w1.z = v0.w; w1.w = v1.w;
            float* q = &ldsB[j * LPAIR + 2 * cc];   // 16B-aligned
            *(float4*)(q)     = w0;          // ds_store_b128
            *(float4*)(q + 4) = w1;          // ds_store_b128
        }
    }
    __syncthreads();

    const int wave = tid >> 5;
    const int lane = tid & 31;
    const int half = lane >> 4;              // 0: lanes 0-15, 1: lanes 16-31
    const int l16  = lane & 15;
    const int rowBase = blockIdx.x * 64 + wave * 16;
    if (rowBase >= M) return;                // wave-uniform; EXEC all-1s for WMMA

    v8f acc[4];
    v8f zero = {};
#pragma unroll
    for (int t = 0; t < 4; ++t) acc[t] = zero;

    const float* Arow   = A + (size_t)(rowBase + l16) * K + 2 * half;
    const float* bpLane = &ldsB[half * LPAIR + 2 * l16];   // row-pair (k/2 + half), col l16
    v2f a_next = *(const v2f*)(Arow);        // k = 0 fragment
#pragma unroll
    for (int k = 0; k < K; k += 4) {
        const v2f a = a_next;
        if (k + 4 < K) a_next = *(const v2f*)(Arow + k + 4);   // prefetch next step
        const float* bp = bpLane + (k >> 1) * LPAIR;           // compile-time offset
#pragma unroll
        for (int t = 0; t < 4; ++t) {
            const v2f b = *(const v2f*)(bp + 32 * t);          // one ds_load_b64
            acc[t] = __builtin_amdgcn_wmma_f32_16x16x4_f32(
                false, a, false, b, (short)0, acc[t], false, false);
        }
    }

#pragma unroll
    for (int t = 0; t < 4; ++t) {
        const int col = colBase + t * 16 + l16;
        const float bv = bias ? bias[col] : 0.0f;
#pragma unroll
        for (int r = 0; r < 8; ++r) {
            const int row = rowBase + r + 8 * half;   // 16x16 f32 C/D layout (CDNA5)
            C[(size_t)row * ncols + col] = acc[t][r] + bv;
        }
    }
}

// ---------------- attention ----------------
// alpha_s[n,h] = <h[n,h,:], a_src[h,:]>,  alpha_d likewise
__global__ void node_alpha(const float* __restrict__ h, const float* __restrict__ a_src,
                           const float* __restrict__ a_dst, float* __restrict__ as_out,
                           float* __restrict__ ad_out) {
    int gid = blockIdx.x * blockDim.x + threadIdx.x;
    if (gid >= NN * HEADS) return;
    int n = gid >> 2, hd = gid & 3;
    const float* hp = h + (size_t)n * HCD + hd * CHD;
    const float* sp = a_src + hd * CHD;
    const float* dp = a_dst + hd * CHD;
    float s = 0.f, d = 0.f;
#pragma unroll 8
    for (int c = 0; c < CHD; ++c) {
        float v = hp[c];
        s += v * sp[c];
        d += v * dp[c];
    }
    as_out[gid] = s;
    ad_out[gid] = d;
}

// pass 1: e = leaky_relu(as[src]+ad[dst]); store; atomic-max per (dst,head)
__global__ void edge_pass1(const int* __restrict__ ei, const float* __restrict__ as_,
                           const float* __restrict__ ad_, float* __restrict__ epass,
                           float* __restrict__ emax) {
    int gid = blockIdx.x * blockDim.x + threadIdx.x;
    if (gid >= ETOT * HEADS) return;
    int e = gid >> 2, hd = gid & 3;
    int s, d; edge_sd(ei, e, s, d);
    float v = as_[s * HEADS + hd] + ad_[d * HEADS + hd];
    v = v >= 0.f ? v : NSLOPE * v;
    epass[gid] = v;
    atomicMaxF(&emax[d * HEADS + hd], v);
}

// pass 2: ex = exp(e - emax[dst]); store; atomic denom sum per (dst,head)
__global__ void edge_pass2(const int* __restrict__ ei, float* __restrict__ epass,
                           const float* __restrict__ emax, float* __restrict__ denomB) {
    int gid = blockIdx.x * blockDim.x + threadIdx.x;
    if (gid >= ETOT * HEADS) return;
    int e = gid >> 2, hd = gid & 3;
    int s, d; edge_sd(ei, e, s, d);
    float m = emax[d * HEADS + hd];
    if (!isfinite(m)) m = 0.0f;
    float ex = expf(epass[gid] - m);
    epass[gid] = ex;
    atomicAdd(&denomB[d * HEADS + hd], ex);
}

// pass 3: one wave per edge, 8 channels/lane; scatter alpha*h[src] into hagg[dst]
__global__ void edge_agg(const int* __restrict__ ei, const float* __restrict__ epass,
                         const float* __restrict__ denomB, const float* __restrict__ hfeat,
                         float* __restrict__ hagg) {
    int wid = (blockIdx.x * blockDim.x + threadIdx.x) >> 5;   // wave-uniform
    if (wid >= ETOT) return;
    int lane = threadIdx.x & 31;
    int s, d; edge_sd(ei, wid, s, d);
    const float* hs = hfeat + (size_t)s * HCD;
    float* ha = hagg + (size_t)d * HCD;
#pragma unroll
    for (int j = 0; j < 8; ++j) {
        int c = j * 32 + lane;
        int hd = c >> 6;
        float alpha = epass[wid * HEADS + hd] / denomB[d * HEADS + hd];
        atomicAdd(&ha[c], hs[c] * alpha);
    }
}

// ---------------- batchnorm + elu ----------------
#define BN_ROWS 128
__global__ void bn_partial(const float* __restrict__ h, float* __restrict__ bn_sum,
                           float* __restrict__ bn_sumsq) {
    int c = threadIdx.x;                  // 256 threads = 256 channels (coalesced)
    int r0 = blockIdx.x * BN_ROWS;
    float s = 0.f, s2 = 0.f;
    for (int r = 0; r < BN_ROWS; ++r) {
        int row = r0 + r;
        if (row < NN) {
            float v = h[(size_t)row * HCD + c];
            s += v; s2 += v * v;
        }
    }
    atomicAdd(&bn_sum[c], s);
    atomicAdd(&bn_sumsq[c], s2);
}

__global__ void bn_finalize(const float* __restrict__ bn_sum, const float* __restrict__ bn_sumsq,
                            float* __restrict__ bn_mean, float* __restrict__ bn_rstd) {
    int c = threadIdx.x;
    float m = bn_sum[c] / (float)NN;
    float v = bn_sumsq[c] / (float)NN - m * m;
    bn_mean[c] = m;
    bn_rstd[c] = rsqrtf(v + EPSBN);
}

__global__ void bn_elu_apply(float* __restrict__ h, const float* __restrict__ bn_mean,
                             const float* __restrict__ bn_rstd, const float* __restrict__ gamma,
                             const float* __restrict__ beta) {
    int gid = blockIdx.x * blockDim.x + threadIdx.x;
    if (gid >= NN * HCD) return;
    int c = gid & (HCD - 1);
    float v = gamma[c] * (h[gid] - bn_mean[c]) * bn_rstd[c] + beta[c];
    h[gid] = v > 0.f ? v : expm1f(v);
}

// ---------------- pooling ----------------
__global__ void pool_partial(const float* __restrict__ h, const int* __restrict__ batch,
                             float* __restrict__ pooled) {
    int gid = blockIdx.x * blockDim.x + threadIdx.x;
    if (gid >= NN * HCD) return;
    int n = gid >> 8, c = gid & (HCD - 1);
    atomicAdd(&pooled[batch[n] * HCD + c], h[gid]);
}

__global__ void pool_div(float* __restrict__ pooled, const int* __restrict__ counts) {
    int gid = blockIdx.x * blockDim.x + threadIdx.x;
    if (gid >= BB * HCD) return;
    pooled[gid] /= (float)counts[gid >> 8];
}

// ---------------- host side ----------------
static inline size_t alignup(size_t x) { return (x + 255) & ~(size_t)255; }
static inline int cdiv(long a, long b) { return (int)((a + b - 1) / b); }

static void run_gat_layer_common(const float* a_src, const float* a_dst,
                                 const float* gamma, const float* beta,
                                 const int* ei,
                                 const float* hfeat, float* hagg, float* epass,
                                 float* as_, float* ad_, float* emax, float* denomB,
                                 float* bn_sum, float* bn_sumsq, float* bn_mean, float* bn_rstd,
                                 hipStream_t stream) {
    node_alpha<<<cdiv((long)NN * HEADS, 256), 256, 0, stream>>>(hfeat, a_src, a_dst, as_, ad_);
    fill_f32<<<cdiv((long)NN * HEADS, 256), 256, 0, stream>>>(emax, -INFINITY, NN * HEADS);
    fill_f32<<<cdiv((long)NN * HEADS, 256), 256, 0, stream>>>(denomB, 0.0f, NN * HEADS);
    fill_f32<<<cdiv((long)NN * HCD, 256), 256, 0, stream>>>(hagg, 0.0f, NN * HCD);
    edge_pass1<<<cdiv((long)ETOT * HEADS, 256), 256, 0, stream>>>(ei, as_, ad_, epass, emax);
    edge_pass2<<<cdiv((long)ETOT * HEADS, 256), 256, 0, stream>>>(ei, epass, emax, denomB);
    edge_agg<<<cdiv((long)ETOT * 32, 256), 256, 0, stream>>>(ei, epass, denomB, hfeat, hagg);
    fill_f32<<<cdiv(2 * HCD, 256), 256, 0, stream>>>(bn_sum, 0.0f, 2 * HCD);
    bn_partial<<<cdiv(NN, BN_ROWS), HCD, 0, stream>>>(hagg, bn_sum, bn_sumsq);
    bn_finalize<<<1, HCD, 0, stream>>>(bn_sum, bn_sumsq, bn_mean, bn_rstd);
    bn_elu_apply<<<cdiv((long)NN * HCD, 256), 256, 0, stream>>>(hagg, bn_mean, bn_rstd, gamma, beta);
}

extern "C" void kernel_launch(void* const* d_in, const int* in_sizes, int n_in,
                              void* d_out, int out_size, void* d_ws, size_t ws_size,
                              hipStream_t stream) {
    // inputs in setup_inputs() order
    const float* x      = (const float*)d_in[0];
    const int*   ei     = (const int*)d_in[1];      // [2,E]: src = ei[0..E), dst = ei[E..2E)
    const int*   batch  = (const int*)d_in[2];
    const float* W_pos  = (const float*)d_in[3];
    const float* b_pos  = (const float*)d_in[4];
    const float* W1     = (const float*)d_in[5];
    const float* a_src1 = (const float*)d_in[6];
    const float* a_dst1 = (const float*)d_in[7];
    const float* gamma1 = (const float*)d_in[9];
    const float* beta1  = (const float*)d_in[10];
    const float* W2     = (const float*)d_in[11];
    const float* a_src2 = (const float*)d_in[12];
    const float* a_dst2 = (const float*)d_in[13];
    const float* gamma2 = (const float*)d_in[15];
    const float* beta2  = (const float*)d_in[16];
    const float* W_fc   = (const float*)d_in[17];
    const float* b_fc   = (const float*)d_in[18];
    float* out = (float*)d_out;                      // [B, OUT]

    // ---- carve workspace ----
    char* p = (char*)d_ws;
    float* h0     = (float*)p; p += alignup(sizeof(float) * (size_t)NN * ENHD);
    float* hfeat  = (float*)p; p += alignup(sizeof(float) * (size_t)NN * HCD);
    float* hagg   = (float*)p; p += alignup(sizeof(float) * (size_t)NN * HCD);
    float* epass  = (float*)p; p += alignup(sizeof(float) * (size_t)ETOT * HEADS);
    float* as_    = (float*)p; p += alignup(sizeof(float) * (size_t)NN * HEADS);
    float* ad_    = (float*)p; p += alignup(sizeof(float) * (size_t)NN * HEADS);
    float* emax   = (float*)p; p += alignup(sizeof(float) * (size_t)NN * HEADS);
    float* denomB = (float*)p; p += alignup(sizeof(float) * (size_t)NN * HEADS);
    float* bn_sum = (float*)p; p += alignup(sizeof(float) * (size_t)2 * HCD);  // sum | sumsq
    float* bn_sumsq = bn_sum + HCD;
    float* bn_mean  = (float*)p; p += alignup(sizeof(float) * HCD);
    float* bn_rstd  = (float*)p; p += alignup(sizeof(float) * HCD);
    float* pooled   = (float*)p; p += alignup(sizeof(float) * (size_t)BB * HCD);
    int* counts = (int*)p; p += alignup(sizeof(int) * BB);
    int* starts = (int*)p; p += alignup(sizeof(int) * BB);
    int* gsz    = (int*)p; p += alignup(sizeof(int) * BB);
    (void)ws_size; (void)in_sizes; (void)n_in; (void)out_size;

    // ---- graph meta + positional encoding ----
    zero_i32<<<1, BB, 0, stream>>>(counts, BB);
    count_nodes<<<cdiv(NN, 256), 256, 0, stream>>>(batch, counts);
    graph_meta<<<1, 1, 0, stream>>>(counts, starts, gsz);
    build_h0<<<cdiv((long)NN * ENHD, 256), 256, 0, stream>>>(x, batch, starts, gsz,
                                                             W_pos, b_pos, h0);

    dim3 gridN(cdiv(NN, 64), HCD / 64);

    // ---- layer 1: hfeat = h0 @ W1 (K=144, WMMA) -> attention -> hagg ----
    gemm_wmma_f32_t<ENHD><<<gridN, 128, 0, stream>>>(h0, W1, nullptr, hfeat, NN, HCD);
    run_gat_layer_common(a_src1, a_dst1, gamma1, beta1, ei,
                         hfeat, hagg, epass, as_, ad_, emax, denomB,
                         bn_sum, bn_sumsq, bn_mean, bn_rstd, stream);

    // ---- layer 2: hfeat = hagg @ W2 (K=256, WMMA); hagg reused as agg target ----
    gemm_wmma_f32_t<HCD><<<gridN, 128, 0, stream>>>(hagg, W2, nullptr, hfeat, NN, HCD);
    run_gat_layer_common(a_src2, a_dst2, gamma2, beta2, ei,
                         hfeat, hagg, epass, as_, ad_, emax, denomB,
                         bn_sum, bn_sumsq, bn_mean, bn_rstd, stream);

    // ---- global mean pool + final linear (WMMA, K=256) ----
    fill_f32<<<cdiv((long)BB * HCD, 256), 256, 0, stream>>>(pooled, 0.0f, BB * HCD);
    pool_partial<<<cdiv((long)NN * HCD, 256), 256, 0, stream>>>(hagg, batch, pooled);
    pool_div<<<cdiv((long)BB * HCD, 256), 256, 0, stream>>>(pooled, counts);
    dim3 gf(cdiv(BB, 64), OUTD / 64);
    gemm_wmma_f32_t<HCD><<<gf, 128, 0, stream>>>(pooled, W_fc, b_fc, out, BB, OUTD);
}